// Critic_TS_5892695130619
// MI455X (gfx1250) — compile-verified
//
#include <hip/hip_runtime.h>
#include <math.h>

typedef __attribute__((ext_vector_type(2))) float v2f;
typedef __attribute__((ext_vector_type(8))) float v8f;

#define BATCH 256
#define HWPIX 4096
#define EPS   1e-5f

// d_ws float layout
#define WS_P1    0        // 256*6   bn1 partial sums/sumsq
#define WS_AFF   1600     // 6       bn1 folded affine: alpha[3], delta[3]
#define WS_P2    1664     // 256*9   a-stat partials
#define WS_T     4096     // 128     bn2 folded bias t[g]
#define WS_WC    4224     // 128*4   k_g * w1[g,c] (K padded to 4)
#define WS_STATS 4736     // 256*1152  9 pooled stats per (b,g')
#define WS_WF    299648   // 1152*128  folded conv2 weights (transposed)
#define WS_POOL  447104   // 256*128
#define WS_XR    479872   // 256*128  relu(bn3(pooled))

__device__ __forceinline__ float relu_(float x) { return x > 0.f ? x : 0.f; }

__device__ inline float wave_red(float v) {
  #pragma unroll
  for (int o = 16; o > 0; o >>= 1) v += __shfl_xor(v, o, 32);
  return v;
}

// ---- Pass 1: per-channel sum/sumsq of world_state (bn1 batch stats) ----
__global__ void k_stats1(const float* __restrict__ wsrc, float* __restrict__ ws) {
  __shared__ float red[8][6];
  float s[3] = {0,0,0}, q[3] = {0,0,0};
  const long total = (long)BATCH * 3 * HWPIX;
  for (long i = (long)blockIdx.x * blockDim.x + threadIdx.x; i < total;
       i += (long)gridDim.x * blockDim.x) {
    int c = (int)((i >> 12) % 3);
    float v = wsrc[i];
    s[c] += v; q[c] += v * v;
  }
  int lane = threadIdx.x & 31, wv = threadIdx.x >> 5;
  #pragma unroll
  for (int c = 0; c < 3; c++) { s[c] = wave_red(s[c]); q[c] = wave_red(q[c]); }
  if (lane == 0) {
    red[wv][0]=s[0]; red[wv][1]=s[1]; red[wv][2]=s[2];
    red[wv][3]=q[0]; red[wv][4]=q[1]; red[wv][5]=q[2];
  }
  __syncthreads();
  if (threadIdx.x < 6) {
    float t = 0;
    for (int w = 0; w < 8; w++) t += red[w][threadIdx.x];
    ws[WS_P1 + blockIdx.x * 6 + threadIdx.x] = t;
  }
}

// ---- fold bn1 into a_c = relu(alpha_c * x + delta_c) ----
__global__ void k_prep1(const float* __restrict__ g1, const float* __restrict__ b1,
                        float* __restrict__ ws) {
  __shared__ float tot[6];
  if (threadIdx.x < 6) {
    float t = 0;
    for (int i = 0; i < 256; i++) t += ws[WS_P1 + i * 6 + threadIdx.x];
    tot[threadIdx.x] = t;
  }
  __syncthreads();
  if (threadIdx.x < 3) {
    int c = threadIdx.x;
    const float N = 1048576.f;
    float m = tot[c] / N;
    float var = tot[3 + c] / N - m * m;
    float inv = rsqrtf(var + EPS);
    ws[WS_AFF + c]     = g1[c] * inv;
    ws[WS_AFF + 3 + c] = b1[c] - g1[c] * m * inv;
  }
}

// ---- Pass 2: moments of a = relu(bn1(ws)): sums + 3x3 second moments ----
__global__ void k_stats2(const float* __restrict__ wsrc, float* __restrict__ ws) {
  __shared__ float red[8][9];
  float al0=ws[WS_AFF+0], al1=ws[WS_AFF+1], al2=ws[WS_AFF+2];
  float de0=ws[WS_AFF+3], de1=ws[WS_AFF+4], de2=ws[WS_AFF+5];
  float acc[9] = {0,0,0,0,0,0,0,0,0};
  const long total = (long)BATCH * HWPIX;
  for (long i = (long)blockIdx.x * blockDim.x + threadIdx.x; i < total;
       i += (long)gridDim.x * blockDim.x) {
    long b = i >> 12; int p = (int)(i & 4095);
    long base = b * 12288 + p;
    float a0 = relu_(al0 * wsrc[base]        + de0);
    float a1 = relu_(al1 * wsrc[base + 4096] + de1);
    float a2 = relu_(al2 * wsrc[base + 8192] + de2);
    acc[0]+=a0; acc[1]+=a1; acc[2]+=a2;
    acc[3]+=a0*a0; acc[4]+=a1*a1; acc[5]+=a2*a2;
    acc[6]+=a0*a1; acc[7]+=a0*a2; acc[8]+=a1*a2;
  }
  int lane = threadIdx.x & 31, wv = threadIdx.x >> 5;
  #pragma unroll
  for (int k = 0; k < 9; k++) acc[k] = wave_red(acc[k]);
  if (lane == 0) { for (int k = 0; k < 9; k++) red[wv][k] = acc[k]; }
  __syncthreads();
  if (threadIdx.x < 9) {
    float t = 0;
    for (int w = 0; w < 8; w++) t += red[w][threadIdx.x];
    ws[WS_P2 + blockIdx.x * 9 + threadIdx.x] = t;
  }
}

// ---- per-g bn2 fold: Wc[g][c] = k_g * w1[g,c], t[g] ----
__global__ void k_prep2(const float* __restrict__ w1p, const float* __restrict__ g2,
                        const float* __restrict__ b2, float* __restrict__ ws) {
  __shared__ float st[9];
  if (threadIdx.x < 9) {
    float t = 0;
    for (int i = 0; i < 256; i++) t += ws[WS_P2 + i * 9 + threadIdx.x];
    st[threadIdx.x] = t;
  }
  __syncthreads();
  int g = threadIdx.x;
  const float N = 1048576.f;
  float m0=st[0]/N, m1=st[1]/N, m2=st[2]/N;
  float E00=st[3]/N, E11=st[4]/N, E22=st[5]/N, E01=st[6]/N, E02=st[7]/N, E12=st[8]/N;
  float w0 = w1p[g*3+0], w1 = w1p[g*3+1], w2v = w1p[g*3+2];
  float mu = w0*m0 + w1*m1 + w2v*m2;
  float e2 = w0*w0*E00 + w1*w1*E11 + w2v*w2v*E22
           + 2.f*(w0*w1*E01 + w0*w2v*E02 + w1*w2v*E12);
  float inv = rsqrtf(e2 - mu*mu + EPS);
  float kk = g2[g] * inv;
  ws[WS_T + g] = b2[g] - g2[g] * mu * inv;
  ws[WS_WC + g*4 + 0] = kk * w0;
  ws[WS_WC + g*4 + 1] = kk * w1;
  ws[WS_WC + g*4 + 2] = kk * w2v;
  ws[WS_WC + g*4 + 3] = 0.f;
}

// ---- one 16-pixel x 16-channel tile: WMMA + relu + static edge bookkeeping ----
// No predicated loads: xoff selects the a-plane by address (hoisted), ymask zeroes
// the K=3 lanes by value (post-relu data is finite, so *0 is exact).
template<bool R0, bool RH, bool C0, bool CW>
__device__ __forceinline__ void tile_step(
    const float* __restrict__ alds, int pb, int xoff, float ymask, bool hi,
    v2f bv, const v8f& cm,
    float& pT, float& pr0, float& prH, float& pc0, float& pcW,
    float& pe00, float& pe0W, float& peH0, float& peHW)
{
  v2f av;                            // A (16x4): lanes<16 (a0,a1), lanes>=16 (a2,0)
  av.x = alds[xoff + pb];
  av.y = alds[4096 + pb] * ymask;
  v8f d = __builtin_amdgcn_wmma_f32_16x16x4_f32(false, av, false, bv,
                                                (short)0, cm, false, false);
  float r0 = relu_(d[0]), r1 = relu_(d[1]), r2 = relu_(d[2]), r3 = relu_(d[3]);
  float r4 = relu_(d[4]), r5 = relu_(d[5]), r6 = relu_(d[6]), r7 = relu_(d[7]);
  float s8 = ((r0 + r1) + (r2 + r3)) + ((r4 + r5) + (r6 + r7));
  pT += s8;
  if (R0) pr0 += s8;
  if (RH) prH += s8;
  if (C0) {                          // image col 0 lives in pixel m=0 -> lanes<16, r0
    float e0 = hi ? 0.f : r0;
    pc0 += e0;
    if (R0) pe00 += e0;
    if (RH) peH0 += e0;
  }
  if (CW) {                          // image col 63 lives in pixel m=15 -> lanes>=16, r7
    float eW = hi ? r7 : 0.f;
    pcW += eW;
    if (R0) pe0W += eW;
    if (RH) peHW += eW;
  }
}

// ---- Main: per image, WMMA (16 pix x K4 x 16 ch), relu, 9 edge-aware stats ----
__global__ void __launch_bounds__(256) k_main(const float* __restrict__ wsrc,
                                              float* __restrict__ ws) {
  __shared__ float alds[3 * HWPIX];   // 48 KB: bn1+relu pre-applied image
  int b = blockIdx.x, tid = threadIdx.x;
  float al0=ws[WS_AFF+0], al1=ws[WS_AFF+1], al2=ws[WS_AFF+2];
  float de0=ws[WS_AFF+3], de1=ws[WS_AFF+4], de2=ws[WS_AFF+5];
  const float* img = wsrc + (long)b * 12288;
  for (int i = tid; i < HWPIX; i += 256) {
    alds[i]        = relu_(al0 * img[i]        + de0);
    alds[4096 + i] = relu_(al1 * img[4096 + i] + de1);
    alds[8192 + i] = relu_(al2 * img[8192 + i] + de2);
  }
  __syncthreads();

  int lane = tid & 31, wv = tid >> 5;
  int ln16 = lane & 15;
  bool hi = lane >= 16;
  int g = wv * 16 + ln16;            // this lane's output channel (N = lane%16)
  int xoff = hi ? 8192 : 0;          // hoisted A-plane select (a0 vs a2)
  float ymask = hi ? 0.f : 1.f;      // hoisted K=3 zeroing

  v2f bv;                            // B (4x16): lanes<16 hold K0/K1, lanes>=16 K2/K3
  bv.x = hi ? ws[WS_WC + g*4 + 2] : ws[WS_WC + g*4 + 0];
  bv.y = hi ? 0.f                 : ws[WS_WC + g*4 + 1];
  float tval = ws[WS_T + g];
  const v8f cm = {tval,tval,tval,tval,tval,tval,tval,tval};

  float pT=0, pr0=0, prH=0, pc0=0, pcW=0, pe00=0, pe0W=0, peH0=0, peHW=0;

  // row 0 (tiles 0..3): ROW0 flag statically true
  tile_step<true,false,true ,false>(alds, ln16     , xoff, ymask, hi, bv, cm, pT,pr0,prH,pc0,pcW,pe00,pe0W,peH0,peHW);
  tile_step<true,false,false,false>(alds, ln16 + 16, xoff, ymask, hi, bv, cm, pT,pr0,prH,pc0,pcW,pe00,pe0W,peH0,peHW);
  tile_step<true,false,false,false>(alds, ln16 + 32, xoff, ymask, hi, bv, cm, pT,pr0,prH,pc0,pcW,pe00,pe0W,peH0,peHW);
  tile_step<true,false,false,true >(alds, ln16 + 48, xoff, ymask, hi, bv, cm, pT,pr0,prH,pc0,pcW,pe00,pe0W,peH0,peHW);
  // rows 1..62: no row flags, only static column flags
  for (int row = 1; row < 63; row++) {
    int pb = (row << 6) + ln16;
    tile_step<false,false,true ,false>(alds, pb     , xoff, ymask, hi, bv, cm, pT,pr0,prH,pc0,pcW,pe00,pe0W,peH0,peHW);
    tile_step<false,false,false,false>(alds, pb + 16, xoff, ymask, hi, bv, cm, pT,pr0,prH,pc0,pcW,pe00,pe0W,peH0,peHW);
    tile_step<false,false,false,false>(alds, pb + 32, xoff, ymask, hi, bv, cm, pT,pr0,prH,pc0,pcW,pe00,pe0W,peH0,peHW);
    tile_step<false,false,false,true >(alds, pb + 48, xoff, ymask, hi, bv, cm, pT,pr0,prH,pc0,pcW,pe00,pe0W,peH0,peHW);
  }
  // row 63 (tiles 252..255): ROWH flag statically true
  tile_step<false,true,true ,false>(alds, 4032 + ln16, xoff, ymask, hi, bv, cm, pT,pr0,prH,pc0,pcW,pe00,pe0W,peH0,peHW);
  tile_step<false,true,false,false>(alds, 4048 + ln16, xoff, ymask, hi, bv, cm, pT,pr0,prH,pc0,pcW,pe00,pe0W,peH0,peHW);
  tile_step<false,true,false,false>(alds, 4064 + ln16, xoff, ymask, hi, bv, cm, pT,pr0,prH,pc0,pcW,pe00,pe0W,peH0,peHW);
  tile_step<false,true,false,true >(alds, 4080 + ln16, xoff, ymask, hi, bv, cm, pT,pr0,prH,pc0,pcW,pe00,pe0W,peH0,peHW);

  float st[9] = {pT, pr0, prH, pc0, pcW, pe00, pe0W, peH0, peHW};
  #pragma unroll
  for (int k = 0; k < 9; k++) st[k] += __shfl_xor(st[k], 16, 32);
  if (!hi) {
    float* o = ws + WS_STATS + (long)b * 1152 + g * 9;
    #pragma unroll
    for (int k = 0; k < 9; k++) o[k] = st[k];
  }
}

// ---- fold conv2 taps into 9 stat coefficients (incl. 1/(H*W)) ----
__global__ void k_fold2(const float* __restrict__ w2, float* __restrict__ ws) {
  int idx = blockIdx.x * 256 + threadIdx.x;   // 16384 pairs
  int g = idx >> 7, gp = idx & 127;
  const float* w = w2 + (long)(g * 128 + gp) * 9;
  float w0=w[0],w1=w[1],wv2=w[2],w3=w[3],w4=w[4],w5=w[5],w6=w[6],w7=w[7],w8=w[8];
  const float s = 1.f / 4096.f;
  float c[9];
  c[0] = (w0+w1+wv2+w3+w4+w5+w6+w7+w8);   // total
  c[1] = -(w6+w7+w8);                     // row0
  c[2] = -(w0+w1+wv2);                    // row63
  c[3] = -(wv2+w5+w8);                    // col0
  c[4] = -(w0+w3+w6);                     // col63
  c[5] =  w8;                             // (0,0)
  c[6] =  w6;                             // (0,63)
  c[7] =  wv2;                            // (63,0)
  c[8] =  w0;                             // (63,63)
  float* wf = ws + WS_WF;
  #pragma unroll
  for (int k = 0; k < 9; k++) wf[(gp * 9 + k) * 128 + g] = c[k] * s;
}

// ---- pooled[256,128] = stats[256,1152] x Wf[1152,128] via WMMA ----
__global__ void __launch_bounds__(256) k_pool(float* __restrict__ ws) {
  int lane = threadIdx.x & 31, wv = threadIdx.x >> 5;
  int ln16 = lane & 15;
  bool hi = lane >= 16;
  int b0 = blockIdx.x * 16, g0 = wv * 16;
  const float* srow = ws + WS_STATS + (long)(b0 + ln16) * 1152;
  const float* wf = ws + WS_WF;
  int ko = hi ? 2 : 0;
  int gcol = g0 + ln16;
  v8f acc = {0,0,0,0,0,0,0,0};
  for (int kk = 0; kk < 1152; kk += 4) {
    v2f a;  a.x = srow[kk + ko];               a.y = srow[kk + ko + 1];
    v2f bb; bb.x = wf[(kk + ko) * 128 + gcol]; bb.y = wf[(kk + ko + 1) * 128 + gcol];
    acc = __builtin_amdgcn_wmma_f32_16x16x4_f32(false, a, false, bb,
                                                (short)0, acc, false, false);
  }
  float* pool = ws + WS_POOL;
  int rb = b0 + (hi ? 8 : 0);
  #pragma unroll
  for (int i = 0; i < 8; i++) pool[(rb + i) * 128 + gcol] = acc[i];
}

// ---- bn3 (stats over batch) + relu ----
__global__ void k_bn3(const float* __restrict__ g3, const float* __restrict__ b3,
                      float* __restrict__ ws) {
  int g = threadIdx.x;  // 128
  const float* pool = ws + WS_POOL;
  float s = 0, q = 0;
  for (int b = 0; b < 256; b++) { float v = pool[b * 128 + g]; s += v; q += v * v; }
  float m = s / 256.f, var = q / 256.f - m * m, inv = rsqrtf(var + EPS);
  float kk = g3[g] * inv, tt = b3[g] - g3[g] * m * inv;
  float* xr = ws + WS_XR;
  for (int b = 0; b < 256; b++) xr[b * 128 + g] = relu_(kk * pool[b * 128 + g] + tt);
}

// ---- per-row tail: fc1, agent MLP, action MLP, combined head, dist net ----
__global__ void k_final(const float* __restrict__ agent, const float* __restrict__ action,
                        const float* __restrict__ fc1_w, const float* __restrict__ fc1_b,
                        const float* __restrict__ lmg, const float* __restrict__ lmb,
                        const float* __restrict__ mi_w1, const float* __restrict__ mi_w2,
                        const float* __restrict__ e_cr, const float* __restrict__ e_eq,
                        const float* __restrict__ e_nc, const float* __restrict__ e_ns,
                        const float* __restrict__ e_pl,
                        const float* __restrict__ lcg, const float* __restrict__ lcb,
                        const float* __restrict__ ca_w1, const float* __restrict__ ca_b1,
                        const float* __restrict__ ca_w2, const float* __restrict__ ca_b2,
                        const float* __restrict__ lCg, const float* __restrict__ lCb,
                        const float* __restrict__ c_w,
                        const float* __restrict__ ldg, const float* __restrict__ ldb,
                        const float* __restrict__ d_w1, const float* __restrict__ d_w2,
                        const float* __restrict__ ws, float* __restrict__ out) {
  int b = blockIdx.x * 32 + threadIdx.x;
  const float* xr = ws + WS_XR + (long)b * 128;
  float cat[60];
  // fc1
  for (int j = 0; j < 20; j++) {
    float s = fc1_b[j];
    for (int g = 0; g < 128; g++) s += fc1_w[j * 128 + g] * xr[g];
    cat[j] = relu_(s);
  }
  // agent MLP
  const float* ag = agent + b * 28;
  float m = 0; for (int i = 0; i < 28; i++) m += ag[i]; m /= 28.f;
  float q = 0; for (int i = 0; i < 28; i++) { float d = ag[i] - m; q += d * d; }
  float inv = rsqrtf(q / 28.f + EPS);
  float na[28];
  for (int i = 0; i < 28; i++) na[i] = (ag[i] - m) * inv * lmg[i] + lmb[i];
  float h[100];
  for (int j = 0; j < 100; j++) {
    float s = 0; for (int i = 0; i < 28; i++) s += mi_w1[j * 28 + i] * na[i];
    h[j] = relu_(s);
  }
  for (int j = 0; j < 20; j++) {
    float s = 0; for (int i = 0; i < 100; i++) s += mi_w2[j * 100 + i] * h[i];
    cat[20 + j] = relu_(s);
  }
  // action MLP
  const float* ac = action + b * 15;
  int i4=(int)ac[4], i5=(int)ac[5], i9=(int)ac[9], i10=(int)ac[10], i11=(int)ac[11];
  float acts[24];
  acts[0]=ac[0]; acts[1]=ac[1]; acts[2]=ac[2]; acts[3]=ac[3];
  acts[4]=e_cr[i4*3]; acts[5]=e_cr[i4*3+1]; acts[6]=e_cr[i4*3+2];
  acts[7]=e_eq[i5*3]; acts[8]=e_eq[i5*3+1]; acts[9]=e_eq[i5*3+2];
  acts[10]=ac[6]; acts[11]=ac[7]; acts[12]=ac[8];
  acts[13]=e_nc[i9*3]; acts[14]=e_nc[i9*3+1]; acts[15]=e_nc[i9*3+2];
  acts[16]=e_ns[i10*2]; acts[17]=e_ns[i10*2+1];
  acts[18]=e_pl[i11*3]; acts[19]=e_pl[i11*3+1]; acts[20]=e_pl[i11*3+2];
  acts[21]=ac[12]; acts[22]=ac[13]; acts[23]=ac[14];
  float m2 = 0; for (int i = 0; i < 24; i++) m2 += acts[i]; m2 /= 24.f;
  float q2 = 0; for (int i = 0; i < 24; i++) { float d = acts[i] - m2; q2 += d * d; }
  float inv2 = rsqrtf(q2 / 24.f + EPS);
  float na2[24];
  for (int i = 0; i < 24; i++) na2[i] = (acts[i] - m2) * inv2 * lcg[i] + lcb[i];
  for (int j = 0; j < 100; j++) {
    float s = ca_b1[j];
    for (int i = 0; i < 24; i++) s += ca_w1[j * 24 + i] * na2[i];
    h[j] = relu_(s);
  }
  for (int j = 0; j < 20; j++) {
    float s = ca_b2[j];
    for (int i = 0; i < 100; i++) s += ca_w2[j * 100 + i] * h[i];
    cat[40 + j] = relu_(s);
  }
  // combined head
  float m3 = 0; for (int i = 0; i < 60; i++) m3 += cat[i]; m3 /= 60.f;
  float q3 = 0; for (int i = 0; i < 60; i++) { float d = cat[i] - m3; q3 += d * d; }
  float inv3 = rsqrtf(q3 / 60.f + EPS);
  float sc = 0;
  for (int i = 0; i < 60; i++) sc += c_w[i] * ((cat[i] - m3) * inv3 * lCg[i] + lCb[i]);
  out[b] = tanhf(sc);
  // dist net: input is [zeros(28), agent]
  float m4 = 0; for (int i = 0; i < 28; i++) m4 += ag[i]; m4 /= 56.f;
  float q4 = 0; for (int i = 0; i < 28; i++) q4 += ag[i] * ag[i]; q4 /= 56.f;
  float inv4 = rsqrtf(q4 - m4 * m4 + EPS);
  float h3[50];
  for (int j = 0; j < 50; j++) {
    float s = 0;
    for (int i = 0; i < 28; i++)
      s += d_w1[j * 56 + i] * (ldb[i] - m4 * inv4 * ldg[i]);
    for (int i = 0; i < 28; i++)
      s += d_w1[j * 56 + 28 + i] * ((ag[i] - m4) * inv4 * ldg[28 + i] + ldb[28 + i]);
    h3[j] = tanhf(s);
  }
  float sd = 0;
  for (int j = 0; j < 50; j++) sd += d_w2[j] * h3[j];
  out[256 + b] = tanhf(sd);
}

extern "C" void kernel_launch(void* const* d_in, const int* in_sizes, int n_in,
                              void* d_out, int out_size, void* d_ws, size_t ws_size,
                              hipStream_t stream) {
  (void)in_sizes; (void)n_in; (void)out_size; (void)ws_size;
  const float* agent   = (const float*)d_in[0];
  const float* wsrc    = (const float*)d_in[1];
  const float* action  = (const float*)d_in[2];
  const float* bn1_g   = (const float*)d_in[3];
  const float* bn1_b   = (const float*)d_in[4];
  const float* conv1_w = (const float*)d_in[5];
  const float* bn2_g   = (const float*)d_in[6];
  const float* bn2_b   = (const float*)d_in[7];
  const float* conv2_w = (const float*)d_in[8];
  const float* bn3_g   = (const float*)d_in[9];
  const float* bn3_b   = (const float*)d_in[10];
  const float* fc1_w   = (const float*)d_in[11];
  const float* fc1_b   = (const float*)d_in[12];
  const float* ln_mi_g = (const float*)d_in[13];
  const float* ln_mi_b = (const float*)d_in[14];
  const float* mi_w1   = (const float*)d_in[15];
  const float* mi_w2   = (const float*)d_in[16];
  const float* e_cr    = (const float*)d_in[17];
  const float* e_eq    = (const float*)d_in[18];
  const float* e_nc    = (const float*)d_in[19];
  const float* e_ns    = (const float*)d_in[20];
  const float* e_pl    = (const float*)d_in[21];
  const float* ln_ca_g = (const float*)d_in[22];
  const float* ln_ca_b = (const float*)d_in[23];
  const float* ca_w1   = (const float*)d_in[24];
  const float* ca_b1   = (const float*)d_in[25];
  const float* ca_w2   = (const float*)d_in[26];
  const float* ca_b2   = (const float*)d_in[27];
  const float* ln_c_g  = (const float*)d_in[28];
  const float* ln_c_b  = (const float*)d_in[29];
  const float* c_w     = (const float*)d_in[30];
  const float* ln_d_g  = (const float*)d_in[31];
  const float* ln_d_b  = (const float*)d_in[32];
  const float* d_w1    = (const float*)d_in[33];
  const float* d_w2    = (const float*)d_in[34];
  float* ws  = (float*)d_ws;
  float* out = (float*)d_out;

  k_stats1<<<256, 256, 0, stream>>>(wsrc, ws);
  k_prep1 <<<1,   32,  0, stream>>>(bn1_g, bn1_b, ws);
  k_stats2<<<256, 256, 0, stream>>>(wsrc, ws);
  k_prep2 <<<1,   128, 0, stream>>>(conv1_w, bn2_g, bn2_b, ws);
  k_main  <<<256, 256, 0, stream>>>(wsrc, ws);
  k_fold2 <<<64,  256, 0, stream>>>(conv2_w, ws);
  k_pool  <<<16,  256, 0, stream>>>(ws);
  k_bn3   <<<1,   128, 0, stream>>>(bn3_g, bn3_b, ws);
  k_final <<<8,   32,  0, stream>>>(agent, action, fc1_w, fc1_b, ln_mi_g, ln_mi_b,
                                    mi_w1, mi_w2, e_cr, e_eq, e_nc, e_ns, e_pl,
                                    ln_ca_g, ln_ca_b, ca_w1, ca_b1, ca_w2, ca_b2,
                                    ln_c_g, ln_c_b, c_w, ln_d_g, ln_d_b, d_w1, d_w2,
                                    ws, out);
}